// NCELoss_3925600108902
// MI455X (gfx1250) — compile-verified
//
#include <hip/hip_runtime.h>
#include <math.h>

// ---------------------------------------------------------------------------
// NCE loss for MI455X (gfx1250, wave32).
//   noise GEMM (16384x512)x(512x100) via V_WMMA_F32_16X16X4_F32 (exact fp32)
//   A tile staged to LDS via GLOBAL_LOAD_ASYNC_TO_LDS_B128 (ASYNCcnt pipe)
//   target dots via one wave of vector FMAs
//   memory-bound workload (~65-70 MB HBM -> ~3us floor at 23.3 TB/s)
// ---------------------------------------------------------------------------

typedef __attribute__((ext_vector_type(2))) float v2f;
typedef __attribute__((ext_vector_type(8))) float v8f;

#define V_SZ   50257
#define E_SZ   512
#define B_SZ   8
#define L_SZ   2048
#define NR_SZ  100
#define M_TOT  (B_SZ * L_SZ)          // 16384 tokens
#define LDA    516                    // padded LDS row stride (516 % 64 == 4 -> no bank clash)

__device__ __forceinline__ float softplusf(float x) {
    // stable log(1 + e^x)
    return fmaxf(x, 0.0f) + log1pf(expf(-fabsf(x)));
}

__device__ __forceinline__ void wait_asynccnt0() {
#if defined(__has_builtin)
#if __has_builtin(__builtin_amdgcn_s_wait_asynccnt)
    __builtin_amdgcn_s_wait_asynccnt(0);
    return;
#else
    asm volatile("s_wait_asynccnt 0x0" ::: "memory");
    return;
#endif
#else
    asm volatile("s_wait_asynccnt 0x0" ::: "memory");
#endif
}

__global__ void nce_zero_ws(float* ws) {
    if (threadIdx.x == 0 && blockIdx.x == 0) ws[0] = 0.0f;
}

__global__ __launch_bounds__(256)
void nce_main(const float* __restrict__ inp,     // (M_TOT, E)
              const float* __restrict__ emb,     // (V, E)
              const float* __restrict__ bias,    // (V)
              const float* __restrict__ lpn,     // (V) log-prob of noise dist
              const int*   __restrict__ target,  // (M_TOT)
              const int*   __restrict__ nidx,    // (NR)
              float* __restrict__ ws_sum)
{
    __shared__ float ldsA[16 * LDA];   // 16 tokens x 512 f32, padded rows (~33 KB)
    __shared__ float ldsLoss[16];      // per-token loss accumulators

    const int tid  = threadIdx.x;
    const int lane = tid & 31;
    const int wave = tid >> 5;         // 8 waves of 32 (wave32)
    const int m0   = blockIdx.x * 16;  // first token of this M tile

    const float NORM_TERM = logf((float)V_SZ);
    const float LOG_NR    = logf((float)NR_SZ);

    if (tid < 16) ldsLoss[tid] = 0.0f;

    // ---- async stage of A tile (16 x 512) into padded LDS ------------------
    // GLOBAL_LOAD_ASYNC_TO_LDS_B128: no VGPR round-trip, tracked by ASYNCcnt.
    {
        const float* __restrict__ srcBase = inp + (size_t)m0 * E_SZ;
        #pragma unroll
        for (int i = tid; i < 16 * (E_SZ / 4); i += 256) {
            const int row = i >> 7;          // 128 float4 per row
            const int c4  = i & 127;
            const float* src = srcBase + row * E_SZ + c4 * 4;            // 16B aligned
            unsigned dst = (unsigned)(uintptr_t)&ldsA[row * LDA + c4 * 4]; // 16B aligned
            asm volatile("global_load_async_to_lds_b128 %0, %1, off"
                         :: "v"(dst), "v"(src) : "memory");
        }
    }

    // ---- per-wave setup, overlapped with the async copy --------------------
    int   n_valid = 0, ridx = 0, t = 0;
    float colconst = 0.0f;
    const float* Brow = emb;
    const float* Trow = emb;
    if (wave < 7) {
        const int n0 = wave * 16;
        const int n  = n0 + (lane & 15);
        n_valid      = (n < NR_SZ);
        const int nc = n_valid ? n : (NR_SZ - 1);       // clamp: no OOB gather
        ridx = nidx[nc];
        Brow = emb + (size_t)ridx * E_SZ;
        colconst = bias[ridx] - NORM_TERM - lpn[ridx] - LOG_NR;
    } else {
        const int m    = lane & 15;
        const int half = lane >> 4;
        t = target[m0 + m];
        Trow = emb + (size_t)t * E_SZ + half * (E_SZ / 2);
    }

    wait_asynccnt0();                   // our waves' async copies landed in LDS
    __syncthreads();                    // make the whole tile visible to all waves

    if (wave < 7) {
        // ---- one 16(M) x 16(N) tile of the noise GEMM per wave -------------
        // A fragment (16x4 f32): lane&15 = M row, lane>>4 selects K pair {0,1}/{2,3}
        // B fragment (4x16 f32): mirrored layout, N = lane&15
        const int mA   = lane & 15;
        const int koff = (lane >> 4) << 1;        // 0 or 2
        const float* __restrict__ Arow = &ldsA[mA * LDA];

        v8f acc = {0.f, 0.f, 0.f, 0.f, 0.f, 0.f, 0.f, 0.f};
        #pragma unroll 8
        for (int k = 0; k < E_SZ; k += 4) {
            v2f a = *(const v2f*)(Arow + k + koff);   // ds_load_b64
            v2f b = *(const v2f*)(Brow + k + koff);   // global_load_b64 (L2-resident)
            acc = __builtin_amdgcn_wmma_f32_16x16x4_f32(
                false, a, false, b, (short)0, acc, false, false);
        }

        // acc VGPR r = row (r + 8*(lane>=16)), col = wave*16 + (lane&15)
        const float vmask = n_valid ? 1.0f : 0.0f;
        const int mbase = (lane >> 4) << 3;           // 0 or 8
        #pragma unroll
        for (int r = 0; r < 8; ++r) {
            float x   = acc[r] + colconst;            // logit_model - logit_noise - lnNR
            float bce = softplusf(x) * vmask;         // label = 0 for noise slots
            atomicAdd(&ldsLoss[mbase + r], bce);      // ds_add_f32
        }
    } else {
        // ---- wave 7: target-score dot products (2 lanes per token) ---------
        const int m    = lane & 15;
        const int half = lane >> 4;                   // 0 or 1 -> E/2 each
        const float* __restrict__ Arow = &ldsA[m * LDA] + half * (E_SZ / 2);
        float partial = 0.0f;
        #pragma unroll 4
        for (int j = 0; j < E_SZ / 2; j += 4) {
            float4 av = *(const float4*)(Arow + j);
            float4 tv = *(const float4*)(Trow + j);
            partial = fmaf(av.x, tv.x, partial);
            partial = fmaf(av.y, tv.y, partial);
            partial = fmaf(av.z, tv.z, partial);
            partial = fmaf(av.w, tv.w, partial);
        }
        partial += __shfl_xor(partial, 16, 32);       // combine the two halves
        if (half == 0) {
            float x0   = partial + bias[t] - NORM_TERM - lpn[t] - LOG_NR;
            float bce0 = softplusf(x0) - x0;          // label = 1 at slot 0
            atomicAdd(&ldsLoss[m], bce0);
        }
    }
    __syncthreads();

    if (tid == 0) {
        float s = 0.0f;
        #pragma unroll
        for (int i = 0; i < 16; ++i) s += ldsLoss[i];
        atomicAdd(ws_sum, s);                          // global_atomic_add_f32
    }
}

__global__ void nce_finalize(const float* __restrict__ ws, float* __restrict__ out) {
    out[0] = ws[0] * (1.0f / (float)M_TOT);            // mean over B*L
}

extern "C" void kernel_launch(void* const* d_in, const int* in_sizes, int n_in,
                              void* d_out, int out_size, void* d_ws, size_t ws_size,
                              hipStream_t stream) {
    (void)in_sizes; (void)n_in; (void)out_size; (void)ws_size;
    const float* inp    = (const float*)d_in[0];
    const float* emb    = (const float*)d_in[1];
    const float* bias   = (const float*)d_in[2];
    const float* lpn    = (const float*)d_in[3];
    const int*   target = (const int*)d_in[4];
    const int*   nidx   = (const int*)d_in[5];
    float* ws  = (float*)d_ws;
    float* out = (float*)d_out;

    nce_zero_ws<<<1, 64, 0, stream>>>(ws);
    nce_main<<<M_TOT / 16, 256, 0, stream>>>(inp, emb, bias, lpn, target, nidx, ws);
    nce_finalize<<<1, 1, 0, stream>>>(ws, out);
}